// Backward_warp_18176301597221
// MI455X (gfx1250) — compile-verified
//
#include <hip/hip_runtime.h>

// Backward warp (bilinear gather), B=4 C=32 H=512 W=512, fp32.
// Memory-bound (AI ~0.85 flop/byte). Strategy: keep the 134MB input resident
// in MI455X's 192MB L2 (gathers have ~4x tap overlap), stream the output out
// with non-temporal stores so it never evicts input lines, read flow once
// non-temporally. One thread per pixel, weights computed once, reused for all
// 32 channels; taps addressed as base + immediate offsets {0,4,4W,4W+4}.

namespace {

constexpr int B = 4;
constexpr int C = 32;
constexpr int H = 512;
constexpr int W = 512;
constexpr int HW = H * W;

__global__ __launch_bounds__(256) void bwarp_kernel(
    const float* __restrict__ inp,   // [B,C,H,W]
    const float* __restrict__ flow,  // [B,2,H,W]
    float* __restrict__ out)         // [B,C,H,W]
{
    int p = blockIdx.x * blockDim.x + threadIdx.x;   // pixel id in [0, B*H*W)
    if (p >= B * H * W) return;

    const int w = p & (W - 1);
    const int h = (p >> 9) & (H - 1);
    const int b = p >> 18;

    // flow: [B,2,H,W]; read-once -> non-temporal (don't pollute L2)
    const int fbase = b * (2 * HW) + h * W + w;
    const float fx = __builtin_nontemporal_load(flow + fbase);
    const float fy = __builtin_nontemporal_load(flow + fbase + HW);

    // clamp sample position to image (matches reference jnp.clip)
    float x = fminf(fmaxf((float)w + fx, 0.0f), (float)(W - 1));
    float y = fminf(fmaxf((float)h + fy, 0.0f), (float)(H - 1));

    // integer tap corner, shifted so (x0+1, y0+1) is always in-bounds.
    // At the x == W-1 boundary the shifted weights are (0,1), reproducing the
    // reference exactly while keeping a fixed {0,+1,+W,+W+1} tap pattern.
    const float x0f = floorf(x);
    const float y0f = floorf(y);
    int x0 = (int)x0f; x0 = (x0 > W - 2) ? (W - 2) : x0;
    int y0 = (int)y0f; y0 = (y0 > H - 2) ? (H - 2) : y0;

    const float dx = x - (float)x0;   // exact (Sterbenz / same-quantum)
    const float dy = y - (float)y0;
    const float ox = 1.0f - dx;       // == x1f - x of the reference, bit-exact
    const float oy = 1.0f - dy;

    const float w00 = ox * oy;   // (x1f-x)(y1f-y)
    const float w10 = ox * dy;   // (x1f-x)(y-y0f)
    const float w01 = dx * oy;   // (x-x0f)(y1f-y)
    const float w11 = dx * dy;   // (x-x0f)(y-y0f)

    const float* __restrict__ src = inp + b * (C * HW) + y0 * W + x0;
    float* __restrict__ dst = out + b * (C * HW) + h * W + w;

    // 8-channel batches: 32 gather loads in flight before the FMA chain,
    // good memory-level parallelism per wave32.
#pragma unroll
    for (int c = 0; c < C; c += 8) {
        float v[8];
#pragma unroll
        for (int u = 0; u < 8; ++u) {
            const float* s = src + (c + u) * HW;
            const float t00 = s[0];
            const float t01 = s[1];
            const float t10 = s[W];
            const float t11 = s[W + 1];
            v[u] = fmaf(w00, t00, fmaf(w01, t01, fmaf(w10, t10, w11 * t11)));
        }
#pragma unroll
        for (int u = 0; u < 8; ++u) {
            // write-once output: NT store keeps input resident in L2
            __builtin_nontemporal_store(v[u], dst + (c + u) * HW);
        }
    }
}

} // namespace

extern "C" void kernel_launch(void* const* d_in, const int* in_sizes, int n_in,
                              void* d_out, int out_size, void* d_ws, size_t ws_size,
                              hipStream_t stream) {
    (void)in_sizes; (void)n_in; (void)d_ws; (void)ws_size; (void)out_size;
    const float* inp  = (const float*)d_in[0];
    const float* flow = (const float*)d_in[1];
    float* out = (float*)d_out;

    const int pixels = B * H * W;           // 1,048,576
    const int block = 256;                  // 8 wave32 waves
    const int grid = (pixels + block - 1) / block;  // 4096
    bwarp_kernel<<<grid, block, 0, stream>>>(inp, flow, out);
}